// BaseReduce_56453050139489
// MI455X (gfx1250) — compile-verified
//
#include <hip/hip_runtime.h>
#include <hip/hip_bf16.h>

#define FDIM 128

typedef float v4f __attribute__((ext_vector_type(4)));

// ---------------------------------------------------------------------------
// CSR path kernels
// ---------------------------------------------------------------------------

// counts (aliased into offsets[0..K)) = 0 ; bmax = -1
__global__ __launch_bounds__(256) void k_init_csr(unsigned* __restrict__ counts,
                                                  int* __restrict__ bmax, int K) {
    int i = blockIdx.x * blockDim.x + threadIdx.x;
    if (i < K) { counts[i] = 0u; bmax[i] = -1; }
}

// histogram of cluster ids + per-cluster max batch (== last-writer since batch sorted)
__global__ __launch_bounds__(256) void k_count(const int* __restrict__ cluster,
                                               const int* __restrict__ batch,
                                               unsigned* __restrict__ counts,
                                               int* __restrict__ bmax, int N) {
    int i = blockIdx.x * blockDim.x + threadIdx.x;
    if (i < N) {
        int c = cluster[i];
        atomicAdd(&counts[c], 1u);
        atomicMax(&bmax[c], batch[i]);
    }
}

// --- parallel 3-phase exclusive scan ---------------------------------------

// Phase 1: each block scans its 1024-chunk in place, writes chunk total.
__global__ __launch_bounds__(1024) void k_scan_blocks(unsigned* __restrict__ offsets,
                                                      unsigned* __restrict__ partials,
                                                      int K) {
    __shared__ unsigned sh[32];
    const int lane = threadIdx.x & 31;
    const int w    = threadIdx.x >> 5;
    const int idx  = blockIdx.x * 1024 + threadIdx.x;

    unsigned v = (idx < K) ? offsets[idx] : 0u;
    unsigned inc = v;                        // inclusive wave32 scan
    #pragma unroll
    for (int d = 1; d < 32; d <<= 1) {
        unsigned t = __shfl_up(inc, d, 32);
        if (lane >= d) inc += t;
    }
    if (lane == 31) sh[w] = inc;
    __syncthreads();
    if (w == 0) {
        unsigned wv = sh[lane];
        unsigned winc = wv;
        #pragma unroll
        for (int d = 1; d < 32; d <<= 1) {
            unsigned t = __shfl_up(winc, d, 32);
            if (lane >= d) winc += t;
        }
        sh[lane] = winc - wv;                // exclusive wave offsets
        if (lane == 31) partials[blockIdx.x] = winc;   // chunk total
    }
    __syncthreads();
    if (idx < K) offsets[idx] = sh[w] + (inc - v);     // exclusive within chunk
}

// Phase 2: one block exclusively scans the (<=1024) chunk totals in place.
__global__ __launch_bounds__(1024) void k_scan_partials(unsigned* __restrict__ partials,
                                                        int nb) {
    __shared__ unsigned sh[32];
    const int lane = threadIdx.x & 31;
    const int w    = threadIdx.x >> 5;
    const int idx  = threadIdx.x;

    unsigned v = (idx < nb) ? partials[idx] : 0u;
    unsigned inc = v;
    #pragma unroll
    for (int d = 1; d < 32; d <<= 1) {
        unsigned t = __shfl_up(inc, d, 32);
        if (lane >= d) inc += t;
    }
    if (lane == 31) sh[w] = inc;
    __syncthreads();
    if (w == 0) {
        unsigned wv = sh[lane];
        unsigned winc = wv;
        #pragma unroll
        for (int d = 1; d < 32; d <<= 1) {
            unsigned t = __shfl_up(winc, d, 32);
            if (lane >= d) winc += t;
        }
        sh[lane] = winc - wv;
    }
    __syncthreads();
    if (idx < nb) partials[idx] = sh[w] + (inc - v);
}

// Phase 3: add chunk bases, seed cursor[], write the N sentinel.
__global__ __launch_bounds__(256) void k_scan_apply(unsigned* __restrict__ offsets,
                                                    unsigned* __restrict__ cursor,
                                                    const unsigned* __restrict__ partials,
                                                    int K, int N) {
    int i = blockIdx.x * blockDim.x + threadIdx.x;
    if (i < K) {
        unsigned o = offsets[i] + partials[i >> 10];
        offsets[i] = o;
        cursor[i]  = o;
    }
    if (i == 0) offsets[K] = (unsigned)N;
}

// slot assignment: node_list is fully overwritten (sum of counts == N)
__global__ __launch_bounds__(256) void k_fill(const int* __restrict__ cluster,
                                              unsigned* __restrict__ cursor,
                                              unsigned* __restrict__ node_list, int N) {
    int i = blockIdx.x * blockDim.x + threadIdx.x;
    if (i < N) {
        unsigned slot = atomicAdd(&cursor[cluster[i]], 1u);
        node_list[slot] = (unsigned)i;
    }
}

// One wave32 per cluster. Lane-parallel gather of up to 32 (node id, scale)
// pairs, then uniform-index shuffles (-> v_readlane, scalar row base) feed
// 4 independent NT b128 row loads per iteration + prefetch of the next 4.
// Accumulate in registers; one NT b128 store per lane. No float atomics.
__global__ __launch_bounds__(256) void k_reduce(const float* __restrict__ x,
                                                const float* __restrict__ s_value,
                                                const unsigned* __restrict__ offsets,
                                                const unsigned* __restrict__ node_list,
                                                const int* __restrict__ bmax,
                                                float* __restrict__ pool,
                                                float* __restrict__ batch_pool, int K) {
    const int lane = threadIdx.x & 31;
    int wid = blockIdx.x * (blockDim.x >> 5) + (threadIdx.x >> 5);
    if (wid >= K) return;
    const int k = __builtin_amdgcn_readfirstlane(wid);

    const unsigned j0 = offsets[k], j1 = offsets[k + 1];
    const size_t loff = (size_t)(lane << 2);
    v4f acc = (v4f)0.0f;

    for (unsigned base = j0; base < j1; base += 32) {
        const unsigned idx = base + (unsigned)lane;
        const bool valid = idx < j1;
        unsigned nl = valid ? node_list[idx] : 0u;   // 32 ids in one vector load
        float    sl = valid ? s_value[nl]   : 0.0f;  // 32 scales, lane-parallel
        const int m = (int)((j1 - base < 32u) ? (j1 - base) : 32u);

        int t = 0;
        for (; t + 3 < m; t += 4) {
            unsigned a0 = __shfl(nl, t + 0, 32); float b0 = __shfl(sl, t + 0, 32);
            unsigned a1 = __shfl(nl, t + 1, 32); float b1 = __shfl(sl, t + 1, 32);
            unsigned a2 = __shfl(nl, t + 2, 32); float b2 = __shfl(sl, t + 2, 32);
            unsigned a3 = __shfl(nl, t + 3, 32); float b3 = __shfl(sl, t + 3, 32);
            // lookahead: prefetch the next 4 rows (ids already in registers;
            // wraps/overruns only re-prefetch a valid row — harmless)
            unsigned p0 = __shfl(nl, (t + 4) & 31, 32);
            unsigned p1 = __shfl(nl, (t + 5) & 31, 32);
            unsigned p2 = __shfl(nl, (t + 6) & 31, 32);
            unsigned p3 = __shfl(nl, (t + 7) & 31, 32);
            __builtin_prefetch(x + (size_t)p0 * FDIM + loff, 0, 1);
            __builtin_prefetch(x + (size_t)p1 * FDIM + loff, 0, 1);
            __builtin_prefetch(x + (size_t)p2 * FDIM + loff, 0, 1);
            __builtin_prefetch(x + (size_t)p3 * FDIM + loff, 0, 1);
            v4f v0 = __builtin_nontemporal_load((const v4f*)(x + (size_t)a0 * FDIM + loff));
            v4f v1 = __builtin_nontemporal_load((const v4f*)(x + (size_t)a1 * FDIM + loff));
            v4f v2 = __builtin_nontemporal_load((const v4f*)(x + (size_t)a2 * FDIM + loff));
            v4f v3 = __builtin_nontemporal_load((const v4f*)(x + (size_t)a3 * FDIM + loff));
            acc += v0 * b0;
            acc += v1 * b1;
            acc += v2 * b2;
            acc += v3 * b3;
        }
        for (; t < m; ++t) {
            unsigned a = __shfl(nl, t, 32); float b = __shfl(sl, t, 32);
            v4f v = __builtin_nontemporal_load((const v4f*)(x + (size_t)a * FDIM + loff));
            acc += v * b;
        }
    }

    __builtin_nontemporal_store(acc, (v4f*)(pool + (size_t)k * FDIM + loff));
    if (lane == 0) {
        int bv = bmax[k];
        batch_pool[k] = (bv < 0) ? (float)k : (float)bv;
    }
}

// ---------------------------------------------------------------------------
// Fallback path (workspace too small): wave-per-row atomic scatter
// ---------------------------------------------------------------------------

__global__ __launch_bounds__(256) void k_init_simple(float* __restrict__ pool,
                                                     float* __restrict__ batch_pool,
                                                     int KF, int K) {
    int i = blockIdx.x * blockDim.x + threadIdx.x;
    if (i < KF) pool[i] = 0.0f;
    if (i < K)  batch_pool[i] = (float)i;
}

__global__ __launch_bounds__(256) void k_scatter(const float* __restrict__ x,
                                                 const float* __restrict__ s_value,
                                                 const int* __restrict__ cluster,
                                                 const int* __restrict__ batch,
                                                 float* __restrict__ pool,
                                                 float* __restrict__ batch_pool, int N) {
    const int lane = threadIdx.x & 31;
    int wid = blockIdx.x * (blockDim.x >> 5) + (threadIdx.x >> 5);
    if (wid >= N) return;
    const int i = __builtin_amdgcn_readfirstlane(wid);

    float sv = s_value[i];
    int c = cluster[i];
    v4f v = __builtin_nontemporal_load((const v4f*)(x + (size_t)i * FDIM + (lane << 2)));
    float* dst = pool + (size_t)c * FDIM + (lane << 2);
    atomicAdd(dst + 0, v.x * sv);
    atomicAdd(dst + 1, v.y * sv);
    atomicAdd(dst + 2, v.z * sv);
    atomicAdd(dst + 3, v.w * sv);
    if (lane == 0) batch_pool[c] = (float)batch[i];
}

// ---------------------------------------------------------------------------

extern "C" void kernel_launch(void* const* d_in, const int* in_sizes, int n_in,
                              void* d_out, int out_size, void* d_ws, size_t ws_size,
                              hipStream_t stream) {
    const float* x        = (const float*)d_in[0];
    const float* s_value  = (const float*)d_in[1];
    // d_in[2] = node_index == arange(N): identity gather, ignored
    const int*   cluster  = (const int*)d_in[3];
    const int*   batch    = (const int*)d_in[4];

    const int N = in_sizes[1];                 // s_value is [N]
    const int K = out_size / (FDIM + 1);       // out = [K*F pool | K batch_pool]
    const int nb = (K + 1023) / 1024;          // scan chunks (<= 1024 supported)

    float* pool       = (float*)d_out;
    float* batch_pool = pool + (size_t)K * FDIM;

    // CSR scratch: offsets[K+1] | cursor[K] | bmax[K] | node_list[N] | partials[nb]
    const size_t need = sizeof(unsigned) * ((size_t)3 * K + 1 + (size_t)N + (size_t)nb);

    if (ws_size >= need && nb <= 1024) {
        unsigned* offsets   = (unsigned*)d_ws;          // doubles as counts[]
        unsigned* cursor    = offsets + (K + 1);
        int*      bmax      = (int*)(cursor + K);
        unsigned* node_list = (unsigned*)(bmax + K);
        unsigned* partials  = node_list + N;

        k_init_csr     <<<(K + 255) / 256, 256, 0, stream>>>(offsets, bmax, K);
        k_count        <<<(N + 255) / 256, 256, 0, stream>>>(cluster, batch, offsets, bmax, N);
        k_scan_blocks  <<<nb, 1024, 0, stream>>>(offsets, partials, K);
        k_scan_partials<<<1, 1024, 0, stream>>>(partials, nb);
        k_scan_apply   <<<(K + 255) / 256, 256, 0, stream>>>(offsets, cursor, partials, K, N);
        k_fill         <<<(N + 255) / 256, 256, 0, stream>>>(cluster, cursor, node_list, N);
        k_reduce       <<<(K + 7) / 8, 256, 0, stream>>>(x, s_value, offsets, node_list,
                                                         bmax, pool, batch_pool, K);
    } else {
        const int KF = K * FDIM;
        k_init_simple<<<(KF + 255) / 256, 256, 0, stream>>>(pool, batch_pool, KF, K);
        k_scatter    <<<(N + 7) / 8, 256, 0, stream>>>(x, s_value, cluster, batch,
                                                       pool, batch_pool, N);
    }
}